// SeqRecModel_24060406792472
// MI455X (gfx1250) — compile-verified
//
#include <hip/hip_runtime.h>
#include <hip/hip_bf16.h>

typedef float v2f __attribute__((ext_vector_type(2)));
typedef float v8f __attribute__((ext_vector_type(8)));

#define DD     128
#define NSESS  1024
#define LP     49
#define MROWS  (NSESS*LP)    // 50176 = 3136*16
#define NITEMS 50000
#define NUSERS 20000
#define NNODES 70000
#define NNZ    800000

__device__ __forceinline__ float sigmoidf_(float x) { return 1.0f / (1.0f + __expf(-x)); }

__device__ __forceinline__ v8f wmma4(v2f a, v2f b, v8f c) {
  return __builtin_amdgcn_wmma_f32_16x16x4_f32(false, a, false, b, (short)0, c, false, false);
}

// ---------------- session prep: lens, target, u_type tile ----------------
__global__ void prep_sessions(const int* __restrict__ ids, const float* __restrict__ u_type,
                              int* __restrict__ lens, float* __restrict__ out_tail) {
  int n = blockIdx.x * blockDim.x + threadIdx.x;
  if (n >= NSESS) return;
  int b = n & 255, s = n >> 8;
  const int* row = ids + (b * 4 + s) * 50;
  int len = 0;
  #pragma unroll
  for (int l = 0; l < 50; ++l) if (row[l] != 0) len = l + 1;
  lens[n] = len;
  out_tail[n]         = (float)row[len - 1];  // target
  out_tail[NSESS + n] = u_type[b];            // tiled u_type
}

// ---------------- gather h = id_emb[seq_in] ----------------
__global__ void gather_h(const int* __restrict__ ids, const int* __restrict__ lens,
                         const float* __restrict__ id_emb, float* __restrict__ sh) {
  int gid = blockIdx.x * blockDim.x + threadIdx.x;
  int r = gid >> 5, lane = gid & 31;
  if (r >= MROWS) return;
  int n = r / LP, l = r % LP;
  int b = n & 255, s = n >> 8;
  int id = (l < lens[n] - 1) ? ids[(b * 4 + s) * 50 + l] : 0;  // id 0 row is zero
  *(float4*)(sh + (size_t)r * DD + lane * 4) =
      *(const float4*)(id_emb + (size_t)id * DD + lane * 4);
}

// ---------------- generic WMMA GEMM: C[Mx128] = A[Mx128] @ W[128x128]^T (+bias) ----------------
template<bool RELU>
__global__ __launch_bounds__(128) void wmma_gemm128(const float* __restrict__ A,
    const float* __restrict__ W, const float* __restrict__ bias,
    float* __restrict__ C, int M) {
  const int wave = blockIdx.x * 4 + (threadIdx.x >> 5);
  const int lane = threadIdx.x & 31;
  const int tm = wave >> 3, tn = wave & 7;
  if (tm * 16 >= M) return;
  const int half = lane >> 4, l16 = lane & 15;
  const float* a = A + (size_t)(tm * 16 + l16) * DD + 2 * half;
  const float* b = W + (size_t)(tn * 16 + l16) * DD + 2 * half;
  v8f acc = {0.f, 0.f, 0.f, 0.f, 0.f, 0.f, 0.f, 0.f};
  #pragma unroll
  for (int k = 0; k < DD; k += 4)
    acc = wmma4(*(const v2f*)(a + k), *(const v2f*)(b + k), acc);
  const float bb = bias ? bias[tn * 16 + l16] : 0.f;
  #pragma unroll
  for (int i = 0; i < 8; ++i) {
    float v = acc[i] + bb;
    if (RELU) v = fmaxf(v, 0.f);
    C[(size_t)(tm * 16 + i + 8 * half) * DD + tn * 16 + l16] = v;
  }
}

// ---------------- HGNN update GEMM with per-row-tile weight select + ReLU ----------------
__global__ __launch_bounds__(128) void wmma_gemm_update(const float* __restrict__ A,
    const float* __restrict__ Wi, const float* __restrict__ Wu,
    const float* __restrict__ bi, const float* __restrict__ bu,
    float* __restrict__ C, int M, int split) {
  const int wave = blockIdx.x * 4 + (threadIdx.x >> 5);
  const int lane = threadIdx.x & 31;
  const int tm = wave >> 3, tn = wave & 7;
  if (tm * 16 >= M) return;
  const bool item = (tm * 16 < split);
  const float* W    = item ? Wi : Wu;
  const float* bias = item ? bi : bu;
  const int half = lane >> 4, l16 = lane & 15;
  const float* a = A + (size_t)(tm * 16 + l16) * DD + 2 * half;
  const float* b = W + (size_t)(tn * 16 + l16) * DD + 2 * half;
  v8f acc = {0.f, 0.f, 0.f, 0.f, 0.f, 0.f, 0.f, 0.f};
  #pragma unroll
  for (int k = 0; k < DD; k += 4)
    acc = wmma4(*(const v2f*)(a + k), *(const v2f*)(b + k), acc);
  const float bb = bias[tn * 16 + l16];
  #pragma unroll
  for (int i = 0; i < 8; ++i)
    C[(size_t)(tm * 16 + i + 8 * half) * DD + tn * 16 + l16] = fmaxf(acc[i] + bb, 0.f);
}

// ---------------- final fuse GEMM: reps = [c_cur | c_hist] @ fuse_w^T + fuse_b (K=256) ----------------
__global__ __launch_bounds__(128) void wmma_gemm_fuse(const float* __restrict__ Ac,
    const float* __restrict__ Ah, const float* __restrict__ W,
    const float* __restrict__ bias, float* __restrict__ C, int M) {
  const int wave = blockIdx.x * 4 + (threadIdx.x >> 5);
  const int lane = threadIdx.x & 31;
  const int tm = wave >> 3, tn = wave & 7;
  if (tm * 16 >= M) return;
  const int half = lane >> 4, l16 = lane & 15;
  const int arow = tm * 16 + l16;
  const float* b = W + (size_t)(tn * 16 + l16) * 256 + 2 * half;
  v8f acc = {0.f, 0.f, 0.f, 0.f, 0.f, 0.f, 0.f, 0.f};
  #pragma unroll
  for (int k = 0; k < 256; k += 4) {
    const float* ap = (k < DD) ? (Ac + (size_t)arow * DD + k)
                               : (Ah + (size_t)arow * DD + (k - DD));
    acc = wmma4(*(const v2f*)(ap + 2 * half), *(const v2f*)(b + k), acc);
  }
  const float bb = bias[tn * 16 + l16];
  #pragma unroll
  for (int i = 0; i < 8; ++i)
    C[(size_t)(tm * 16 + i + 8 * half) * DD + tn * 16 + l16] = acc[i] + bb;
}

// ---------------- constant: add384[j] = b_iah@WihA[j] + b_oah@WihB[j] + b_ih[j] ----------------
__global__ void precompute_add384(const float* __restrict__ w_ih, const float* __restrict__ b_iah,
                                  const float* __restrict__ b_oah, const float* __restrict__ b_ih,
                                  float* __restrict__ add384) {
  int j = blockIdx.x * blockDim.x + threadIdx.x;
  if (j >= 384) return;
  float s = b_ih[j];
  const float* row = w_ih + (size_t)j * 256;
  for (int d = 0; d < DD; ++d) s += b_iah[d] * row[d] + b_oah[d] * row[DD + d];
  add384[j] = s;
}

// ---------------- fused GRU cell: 9 WMMAs/K-step, gate epilogue in registers ----------------
__global__ __launch_bounds__(128) void gru_fused(const float* __restrict__ sh,
    const float* __restrict__ whi, const float* __restrict__ who,
    const float* __restrict__ w_ih, const float* __restrict__ w_hh,
    const float* __restrict__ add384, const float* __restrict__ b_hh,
    const int* __restrict__ lens, float* __restrict__ hnew) {
  const int wave = blockIdx.x * 4 + (threadIdx.x >> 5);
  const int lane = threadIdx.x & 31;
  const int tm = wave >> 3, cb = wave & 7;
  if (tm * 16 >= MROWS) return;
  const int half = lane >> 4, l16 = lane & 15;
  const int r = tm * 16 + l16;
  const int n = r / LP, l = r % LP;
  const int len = lens[n];
  // a_in: m_i[l] = (1<=l<=len-2) * Whi[l-1];  a_out: m_o[l] = (l<=len-3) * Who[l+1]
  const bool in_ok  = (l >= 1) && (l <= len - 2);
  const bool out_ok = (l <= len - 3);
  const float mI = in_ok ? 1.f : 0.f;
  const float mO = out_ok ? 1.f : 0.f;
  const float* aH = sh  + (size_t)r * DD + 2 * half;
  const float* aI = whi + (size_t)(in_ok  ? r - 1 : r) * DD + 2 * half;
  const float* aO = who + (size_t)(out_ok ? r + 1 : r) * DD + 2 * half;
  const int col = cb * 16 + l16;  // B row index == output column
  const float* bR = w_ih + (size_t)(col)       * 256 + 2 * half;
  const float* bZ = w_ih + (size_t)(128 + col) * 256 + 2 * half;
  const float* bN = w_ih + (size_t)(256 + col) * 256 + 2 * half;
  const float* hR = w_hh + (size_t)(col)       * DD + 2 * half;
  const float* hZ = w_hh + (size_t)(128 + col) * DD + 2 * half;
  const float* hN = w_hh + (size_t)(256 + col) * DD + 2 * half;
  v8f aRi = {0.f,0.f,0.f,0.f,0.f,0.f,0.f,0.f}, aZi = aRi, aNi = aRi;
  v8f aRh = aRi, aZh = aRi, aNh = aRi;
  #pragma unroll
  for (int k = 0; k < DD; k += 4) {
    v2f vh = *(const v2f*)(aH + k);
    v2f vi = *(const v2f*)(aI + k); vi.x *= mI; vi.y *= mI;
    v2f vo = *(const v2f*)(aO + k); vo.x *= mO; vo.y *= mO;
    aRi = wmma4(vi, *(const v2f*)(bR + k),      aRi);
    aRi = wmma4(vo, *(const v2f*)(bR + DD + k), aRi);
    aZi = wmma4(vi, *(const v2f*)(bZ + k),      aZi);
    aZi = wmma4(vo, *(const v2f*)(bZ + DD + k), aZi);
    aNi = wmma4(vi, *(const v2f*)(bN + k),      aNi);
    aNi = wmma4(vo, *(const v2f*)(bN + DD + k), aNi);
    aRh = wmma4(vh, *(const v2f*)(hR + k), aRh);
    aZh = wmma4(vh, *(const v2f*)(hZ + k), aZh);
    aNh = wmma4(vh, *(const v2f*)(hN + k), aNh);
  }
  const float aR = add384[col], aZ = add384[128 + col], aN = add384[256 + col];
  const float br = b_hh[col],   bz = b_hh[128 + col],   bn = b_hh[256 + col];
  #pragma unroll
  for (int i = 0; i < 8; ++i) {
    const int rr = tm * 16 + i + 8 * half;
    const float hv = sh[(size_t)rr * DD + col];
    const float rg = sigmoidf_(aRi[i] + aR + aRh[i] + br);
    const float zg = sigmoidf_(aZi[i] + aZ + aZh[i] + bz);
    const float ng = tanhf(aNi[i] + aN + rg * (aNh[i] + bn));
    hnew[(size_t)rr * DD + col] = ng + zg * (hv - ng);
  }
}

// ---------------- gather ht = h[lens-2] ----------------
__global__ void gather_ht(const float* __restrict__ hnew, const int* __restrict__ lens,
                          float* __restrict__ ht) {
  int gid = blockIdx.x * blockDim.x + threadIdx.x;
  int n = gid >> 5, lane = gid & 31;
  if (n >= NSESS) return;
  int r = n * LP + lens[n] - 2;
  *(float4*)(ht + (size_t)n * DD + lane * 4) =
      *(const float4*)(hnew + (size_t)r * DD + lane * 4);
}

// ---------------- attention readout -> c_cur (one wave per session) ----------------
__global__ void attention(const float* __restrict__ q1h, const float* __restrict__ q2h,
                          const float* __restrict__ hnew, const float* __restrict__ att_w,
                          const int* __restrict__ lens, float* __restrict__ ccur) {
  int wave = blockIdx.x * (blockDim.x >> 5) + (threadIdx.x >> 5);
  int lane = threadIdx.x & 31;
  if (wave >= NSESS) return;
  const int n = wave, len = lens[n];
  float q1v[4], aw[4], cacc[4] = {0.f, 0.f, 0.f, 0.f};
  #pragma unroll
  for (int j = 0; j < 4; ++j) {
    q1v[j] = q1h[(size_t)n * DD + lane * 4 + j];
    aw[j]  = att_w[lane * 4 + j];
  }
  float asum = 0.f;
  for (int l = 0; l < len - 1; ++l) {
    const size_t r = (size_t)(n * LP + l) * DD + lane * 4;
    float p = 0.f, hv[4];
    #pragma unroll
    for (int j = 0; j < 4; ++j) {
      hv[j] = hnew[r + j];
      p += sigmoidf_(q1v[j] + q2h[r + j]) * aw[j];
    }
    #pragma unroll
    for (int off = 16; off > 0; off >>= 1) p += __shfl_xor(p, off, 32);
    asum += p;
    #pragma unroll
    for (int j = 0; j < 4; ++j) cacc[j] += p * hv[j];
  }
  const float inv = 1.f / (asum + 1e-8f);
  #pragma unroll
  for (int j = 0; j < 4; ++j) ccur[(size_t)n * DD + lane * 4 + j] = cacc[j] * inv;
}

// ---------------- edge scatter: msg[row] += Wh[col]*val (wave per edge) ----------------
__global__ void scatter_edges(const int* __restrict__ rows, const int* __restrict__ cols,
                              const float* __restrict__ vals, const float* __restrict__ Wh,
                              float* __restrict__ msg, int nnz) {
  int gid = blockIdx.x * blockDim.x + threadIdx.x;
  int e = gid >> 5, lane = gid & 31;
  if (e >= nnz) return;
  const int rr = rows[e], cc = cols[e];
  const float v = vals[e];
  const float4 w = *(const float4*)(Wh + (size_t)cc * DD + lane * 4);
  float* dst = msg + (size_t)rr * DD + lane * 4;
  atomicAdd(dst + 0, w.x * v);
  atomicAdd(dst + 1, w.y * v);
  atomicAdd(dst + 2, w.z * v);
  atomicAdd(dst + 3, w.w * v);
}

// ---------------- msg += h ----------------
__global__ void add_inplace(float* __restrict__ dst, const float* __restrict__ src, int n4) {
  int i = blockIdx.x * blockDim.x + threadIdx.x;
  if (i >= n4) return;
  float4 a = *(float4*)(dst + (size_t)i * 4);
  float4 b = *(const float4*)(src + (size_t)i * 4);
  a.x += b.x; a.y += b.y; a.z += b.z; a.w += b.w;
  *(float4*)(dst + (size_t)i * 4) = a;
}

// ---------------- c_hist = g_user[uid[n%B]] ----------------
__global__ void gather_chist(const float* __restrict__ gh, const int* __restrict__ uid,
                             float* __restrict__ chist) {
  int gid = blockIdx.x * blockDim.x + threadIdx.x;
  int n = gid >> 5, lane = gid & 31;
  if (n >= NSESS) return;
  int u = uid[n & 255];
  *(float4*)(chist + (size_t)n * DD + lane * 4) =
      *(const float4*)(gh + (size_t)(NITEMS + u) * DD + lane * 4);
}

extern "C" void kernel_launch(void* const* d_in, const int* in_sizes, int n_in,
                              void* d_out, int out_size, void* d_ws, size_t ws_size,
                              hipStream_t stream) {
  const float* id_emb   = (const float*)d_in[0];
  const float* user_emb = (const float*)d_in[1];
  const float* lin_r_w  = (const float*)d_in[2];
  const float* upi_w    = (const float*)d_in[3];
  const float* upi_b    = (const float*)d_in[4];
  const float* upu_w    = (const float*)d_in[5];
  const float* upu_b    = (const float*)d_in[6];
  const float* w_ih     = (const float*)d_in[7];
  const float* w_hh     = (const float*)d_in[8];
  const float* b_ih     = (const float*)d_in[9];
  const float* b_hh     = (const float*)d_in[10];
  const float* b_iah    = (const float*)d_in[11];
  const float* b_oah    = (const float*)d_in[12];
  const float* ei_w     = (const float*)d_in[13];
  const float* ei_b     = (const float*)d_in[14];
  const float* eo_w     = (const float*)d_in[15];
  const float* eo_b     = (const float*)d_in[16];
  const float* q1_w     = (const float*)d_in[17];
  const float* q2_w     = (const float*)d_in[18];
  const float* att_w    = (const float*)d_in[19];
  const float* fuse_w   = (const float*)d_in[20];
  const float* fuse_b   = (const float*)d_in[21];
  const int*   rel_rows = (const int*)d_in[22];
  const int*   rel_cols = (const int*)d_in[23];
  const float* rel_vals = (const float*)d_in[24];
  const int*   ids      = (const int*)d_in[25];
  const int*   uid      = (const int*)d_in[26];
  const float* u_type   = (const float*)d_in[27];

  float* out = (float*)d_out;
  float* ws  = (float*)d_ws;

  // workspace layout (floats)
  int*   lens   = (int*)ws;            // 1024
  float* add384 = ws + 1536;           // 384
  float* ht     = ws + 2048;           // 1024*128
  float* q1h    = ht    + NSESS * DD;
  float* ccur   = q1h   + NSESS * DD;
  float* chist  = ccur  + NSESS * DD;
  float* big    = chist + NSESS * DD;
  // session-phase aliases of the big region
  float* sh   = big;
  float* whi  = big + (size_t)1 * MROWS * DD;
  float* who  = big + (size_t)2 * MROWS * DD;
  float* hnew = big + (size_t)3 * MROWS * DD;
  float* q2h  = whi;  // reuse after GRU
  // HGNN-phase aliases (session phase fully done before these are touched)
  float* gh  = big;
  float* wh  = big + (size_t)1 * NNODES * DD;
  float* msg = big + (size_t)2 * NNODES * DD;

  // ---- session phase ----
  prep_sessions<<<4, 256, 0, stream>>>(ids, u_type, lens, out + NSESS * DD);
  gather_h<<<(MROWS * 32) / 256, 256, 0, stream>>>(ids, lens, id_emb, sh);
  wmma_gemm128<false><<<(MROWS / 16) * 8 / 4, 128, 0, stream>>>(sh, ei_w, ei_b, whi, MROWS);
  wmma_gemm128<false><<<(MROWS / 16) * 8 / 4, 128, 0, stream>>>(sh, eo_w, eo_b, who, MROWS);
  precompute_add384<<<2, 256, 0, stream>>>(w_ih, b_iah, b_oah, b_ih, add384);
  gru_fused<<<(MROWS / 16) * 8 / 4, 128, 0, stream>>>(sh, whi, who, w_ih, w_hh,
                                                      add384, b_hh, lens, hnew);
  gather_ht<<<(NSESS * 32) / 256, 256, 0, stream>>>(hnew, lens, ht);
  wmma_gemm128<false><<<(NSESS / 16) * 8 / 4, 128, 0, stream>>>(ht, q1_w, nullptr, q1h, NSESS);
  wmma_gemm128<false><<<(MROWS / 16) * 8 / 4, 128, 0, stream>>>(hnew, q2_w, nullptr, q2h, MROWS);
  attention<<<(NSESS * 32) / 256, 256, 0, stream>>>(q1h, q2h, hnew, att_w, lens, ccur);

  // ---- global HGNN phase (reuses big region) ----
  hipMemcpyAsync(gh, id_emb, (size_t)NITEMS * DD * sizeof(float),
                 hipMemcpyDeviceToDevice, stream);
  hipMemcpyAsync(gh + (size_t)NITEMS * DD, user_emb, (size_t)NUSERS * DD * sizeof(float),
                 hipMemcpyDeviceToDevice, stream);
  for (int k = 0; k < 2; ++k) {
    hipMemsetAsync(msg, 0, (size_t)NNODES * DD * sizeof(float), stream);
    for (int r = 0; r < 3; ++r) {
      wmma_gemm128<false><<<(NNODES / 16) * 8 / 4, 128, 0, stream>>>(
          gh, lin_r_w + (size_t)(k * 3 + r) * DD * DD, nullptr, wh, NNODES);
      scatter_edges<<<(NNZ * 32) / 256, 256, 0, stream>>>(
          rel_rows + (size_t)r * NNZ, rel_cols + (size_t)r * NNZ,
          rel_vals + (size_t)r * NNZ, wh, msg, NNZ);
    }
    add_inplace<<<(NNODES * DD / 4 + 255) / 256, 256, 0, stream>>>(msg, gh, NNODES * DD / 4);
    wmma_gemm_update<<<(NNODES / 16) * 8 / 4, 128, 0, stream>>>(
        msg, upi_w + (size_t)k * DD * DD, upu_w + (size_t)k * DD * DD,
        upi_b + k * DD, upu_b + k * DD, gh, NNODES, NITEMS);
  }

  // ---- fuse ----
  gather_chist<<<(NSESS * 32) / 256, 256, 0, stream>>>(gh, uid, chist);
  wmma_gemm_fuse<<<(NSESS / 16) * 8 / 4, 128, 0, stream>>>(ccur, chist, fuse_w, fuse_b,
                                                           out, NSESS);
}